// QCNNGen323Model_65481071405707
// MI455X (gfx1250) — compile-verified
//
#include <hip/hip_runtime.h>
#include <hip/hip_bf16.h>

typedef __attribute__((ext_vector_type(2))) float v2f;
typedef __attribute__((ext_vector_type(8))) float v8f;

#define BN_EPS 1e-5f
// LDS transpose tile: 16 rows x 18-float stride (even -> b64-aligned reads,
// 18 mod 64-bank pattern -> conflict-free for both write and read patterns).
#define TRS 18
#define TRW (16 * TRS)

// Intra-wave LDS ordering: DS ops of one wave execute in order (ISA 7.3), so
// we only need to stop the compiler from reordering LDS accesses.
__device__ __forceinline__ void wave_sync_lds() {
  __asm__ __volatile__("" ::: "memory");
  __builtin_amdgcn_wave_barrier();
  __asm__ __volatile__("" ::: "memory");
}

__device__ __forceinline__ void bn_coef(const float* __restrict__ sums,
                                        const float* __restrict__ g,
                                        const float* __restrict__ be,
                                        int k, float invB, float& sc, float& sh) {
  const float m = sums[k] * invB;
  const float v = sums[16 + k] * invB - m * m;
  sc = g[k] * rsqrtf(v + BN_EPS);
  sh = be[k] - m * sc;
}

// ---------------------------------------------------------------------------
// Zero BN stat accumulators (5 stages x (sum[16]+sumsq[16])). Runs every
// launch: harness does not re-zero ws between replays.
// ---------------------------------------------------------------------------
__global__ void zero_stats_kernel(float* __restrict__ stats) {
  int i = threadIdx.x;
  if (i < 5 * 32) stats[i] = 0.0f;
}

// ---------------------------------------------------------------------------
// Single-GEMM stage on v_wmma_f32_16x16x4_f32 (optionally BN+ReLU on input,
// optionally storing the pre-BN output). Always accumulates per-feature
// sum/sumsq of the output for the next BatchNorm.
//  A (16x4 f32): lane L<16 -> A[L][0..1]; lane>=16 -> A[L-16][2..3]
//  B (4x16)    : v0 lanes0-15 = B[0][n], lanes16-31 = B[2][n]; v1 rows 1,3
//  C (16x16)   : vgpr r, lane L -> row r+8*(L>>4), col L&15
// ---------------------------------------------------------------------------
template <int KIN, int NOUT, bool BN_IN, bool STORE>
__global__ __launch_bounds__(256)
void stage_kernel(const float* __restrict__ yin,    // [B,KIN]
                  const float* __restrict__ sums0,  // stats of yin (BN_IN)
                  const float* __restrict__ g0,
                  const float* __restrict__ be0,
                  const float* __restrict__ w,      // [NOUT,KIN]
                  const float* __restrict__ bias,   // [NOUT]
                  float* __restrict__ yout,         // [B,NOUT] (STORE)
                  float* __restrict__ sums_out,     // stats accumulator
                  int numTiles, float invB) {
  constexpr int NCH = KIN / 4;
  const int lane = threadIdx.x & 31;
  const int half = lane >> 4;
  const int n    = lane & 15;

  float s0[NCH][2], t0[NCH][2];
  v2f   bw[NCH];
#pragma unroll
  for (int c = 0; c < NCH; ++c) {
    const int kb = c * 4 + 2 * half;
#pragma unroll
    for (int j = 0; j < 2; ++j) {
      if constexpr (BN_IN) bn_coef(sums0, g0, be0, kb + j, invB, s0[c][j], t0[c][j]);
      else { s0[c][j] = 1.0f; t0[c][j] = 0.0f; }
    }
    if (n < NOUT) { bw[c].x = w[n * KIN + kb]; bw[c].y = w[n * KIN + kb + 1]; }
    else          { bw[c].x = 0.0f;            bw[c].y = 0.0f; }
  }
  const float bias_n = (n < NOUT) ? bias[n] : 0.0f;

  const int wavesPerBlock = blockDim.x >> 5;
  const int wave = blockIdx.x * wavesPerBlock + (threadIdx.x >> 5);
  const int nw   = gridDim.x * wavesPerBlock;

  float ssum = 0.0f, ssq = 0.0f;
  for (int tile = wave; tile < numTiles; tile += nw) {
    const size_t rowA = (size_t)tile * 16 + n;
    v8f acc;
#pragma unroll
    for (int r = 0; r < 8; ++r) acc[r] = bias_n;
#pragma unroll
    for (int c = 0; c < NCH; ++c) {
      const int kb = c * 4 + 2 * half;
      const float2 ya = *(const float2*)(yin + rowA * KIN + kb);
      float ax = ya.x, ay = ya.y;
      if constexpr (BN_IN) {
        ax = fmaxf(fmaf(ax, s0[c][0], t0[c][0]), 0.0f);
        ay = fmaxf(fmaf(ay, s0[c][1], t0[c][1]), 0.0f);
      }
      v2f a; a.x = ax; a.y = ay;
      acc = __builtin_amdgcn_wmma_f32_16x16x4_f32(false, a, false, bw[c],
                                                  (short)0, acc, false, false);
    }
    if (n < NOUT) {
      const size_t rbase = (size_t)tile * 16 + 8 * half;
#pragma unroll
      for (int r = 0; r < 8; ++r) {
        const float v = acc[r];
        if constexpr (STORE) yout[(rbase + r) * NOUT + n] = v;
        ssum += v; ssq += v * v;
      }
    }
  }
  ssum += __shfl_xor(ssum, 16, 32);
  ssq  += __shfl_xor(ssq, 16, 32);
  if (half == 0 && n < NOUT) {
    atomicAdd(&sums_out[n], ssum);
    atomicAdd(&sums_out[16 + n], ssq);
  }
}

// ---------------------------------------------------------------------------
// Front chain from the raw input (recompute instead of storing y1/y2):
//   GEMM1: inp[16x8] @ W_fm^T -> y1 (C layout)
//   x1 = relu(BN1(y1));  LDS C->A transpose
//   GEMM2: x1 @ W_c1^T -> y2 (C layout)
//   TAIL=false: accumulate stats of y2, done.
//   TAIL=true : x2 = relu(relu(BN2(y2)) + x1); transpose;
//               GEMM3: x2 @ W_p1^T -> y3; store y3 + stats of y3.
// ---------------------------------------------------------------------------
template <bool TAIL>
__global__ __launch_bounds__(256)
void front_kernel(const float* __restrict__ inp,
                  const float* __restrict__ w0, const float* __restrict__ b0,    // fm [16,8]
                  const float* __restrict__ sums1, const float* __restrict__ g1,
                  const float* __restrict__ be1,
                  const float* __restrict__ w1, const float* __restrict__ b1,    // c1 [16,16]
                  const float* __restrict__ sums2, const float* __restrict__ g2,
                  const float* __restrict__ be2,
                  const float* __restrict__ w2, const float* __restrict__ b2,    // p1 [12,16]
                  float* __restrict__ yout,      // y3 [B,12] (TAIL)
                  float* __restrict__ sums_out,  // stats2 (!TAIL) / stats3 (TAIL)
                  int numTiles, float invB) {
  __shared__ float xls[8 * TRW];
  const int lane = threadIdx.x & 31;
  const int half = lane >> 4;
  const int n    = lane & 15;
  float* myx = &xls[(threadIdx.x >> 5) * TRW];

  v2f bw0[2], bw1[4], bw2[4];
#pragma unroll
  for (int c = 0; c < 2; ++c) {
    const int kb = c * 4 + 2 * half;
    bw0[c].x = w0[n * 8 + kb]; bw0[c].y = w0[n * 8 + kb + 1];
  }
#pragma unroll
  for (int c = 0; c < 4; ++c) {
    const int kb = c * 4 + 2 * half;
    bw1[c].x = w1[n * 16 + kb]; bw1[c].y = w1[n * 16 + kb + 1];
    if constexpr (TAIL) {
      if (n < 12) { bw2[c].x = w2[n * 16 + kb]; bw2[c].y = w2[n * 16 + kb + 1]; }
      else        { bw2[c].x = 0.0f;            bw2[c].y = 0.0f; }
    } else { bw2[c].x = 0.0f; bw2[c].y = 0.0f; }
  }
  const float bias0 = b0[n];
  const float bias1 = b1[n];
  const float bias2 = (TAIL && n < 12) ? b2[n] : 0.0f;

  float s1C, t1C;                       // BN1, C-pattern (per output col n)
  bn_coef(sums1, g1, be1, n, invB, s1C, t1C);
  float s2C = 0.0f, t2C = 0.0f;         // BN2, C-pattern
  if constexpr (TAIL) bn_coef(sums2, g2, be2, n, invB, s2C, t2C);

  const int wavesPerBlock = blockDim.x >> 5;
  const int wave = blockIdx.x * wavesPerBlock + (threadIdx.x >> 5);
  const int nw   = gridDim.x * wavesPerBlock;

  float ssum = 0.0f, ssq = 0.0f;
  for (int tile = wave; tile < numTiles; tile += nw) {
    const size_t rowA = (size_t)tile * 16 + n;

    // --- GEMM1: raw input (no BN) ---
    v8f acc0;
#pragma unroll
    for (int r = 0; r < 8; ++r) acc0[r] = bias0;
#pragma unroll
    for (int c = 0; c < 2; ++c) {
      const int kb = c * 4 + 2 * half;
      const float2 ya = *(const float2*)(inp + rowA * 8 + kb);
      v2f a; a.x = ya.x; a.y = ya.y;
      acc0 = __builtin_amdgcn_wmma_f32_16x16x4_f32(false, a, false, bw0[c],
                                                   (short)0, acc0, false, false);
    }

    // --- x1 = relu(BN1(y1)) in C layout; transpose to A layout ---
    float x1C[8];
#pragma unroll
    for (int r = 0; r < 8; ++r) x1C[r] = fmaxf(fmaf(acc0[r], s1C, t1C), 0.0f);
    wave_sync_lds();
#pragma unroll
    for (int r = 0; r < 8; ++r) myx[(r + 8 * half) * TRS + n] = x1C[r];
    wave_sync_lds();

    // --- GEMM2 ---
    v8f acc1;
#pragma unroll
    for (int r = 0; r < 8; ++r) acc1[r] = bias1;
#pragma unroll
    for (int c = 0; c < 4; ++c) {
      const int kb = c * 4 + 2 * half;
      const float2 xa = *(const float2*)&myx[n * TRS + kb];
      v2f a; a.x = xa.x; a.y = xa.y;
      acc1 = __builtin_amdgcn_wmma_f32_16x16x4_f32(false, a, false, bw1[c],
                                                   (short)0, acc1, false, false);
    }

    if constexpr (!TAIL) {
      // stats of y2 (pre-BN)
#pragma unroll
      for (int r = 0; r < 8; ++r) { ssum += acc1[r]; ssq += acc1[r] * acc1[r]; }
    } else {
      // --- x2 = relu(relu(BN2(y2)) + x1); transpose; GEMM3; store y3 ---
      float x2C[8];
#pragma unroll
      for (int r = 0; r < 8; ++r)
        x2C[r] = fmaxf(fmaxf(fmaf(acc1[r], s2C, t2C), 0.0f) + x1C[r], 0.0f);
      wave_sync_lds();
#pragma unroll
      for (int r = 0; r < 8; ++r) myx[(r + 8 * half) * TRS + n] = x2C[r];
      wave_sync_lds();

      v8f acc2;
#pragma unroll
      for (int r = 0; r < 8; ++r) acc2[r] = bias2;
#pragma unroll
      for (int c = 0; c < 4; ++c) {
        const int kb = c * 4 + 2 * half;
        const float2 xa = *(const float2*)&myx[n * TRS + kb];
        v2f a; a.x = xa.x; a.y = xa.y;
        acc2 = __builtin_amdgcn_wmma_f32_16x16x4_f32(false, a, false, bw2[c],
                                                     (short)0, acc2, false, false);
      }
      if (n < 12) {
        const size_t rbase = (size_t)tile * 16 + 8 * half;
#pragma unroll
        for (int r = 0; r < 8; ++r) {
          const float v = acc2[r];
          yout[(rbase + r) * 12 + n] = v;
          ssum += v; ssq += v * v;
        }
      }
    }
  }

  ssum += __shfl_xor(ssum, 16, 32);
  ssq  += __shfl_xor(ssq, 16, 32);
  const int nstat = TAIL ? 12 : 16;
  if (half == 0 && n < nstat) {
    atomicAdd(&sums_out[n], ssum);
    atomicAdd(&sums_out[16 + n], ssq);
  }
}

// ---------------------------------------------------------------------------
// Back chain from stored y3 (recompute y4 instead of storing it):
//   x3 = relu(BN3(y3))  (loaded in C pattern); transpose
//   GEMM(c2): y4 = x3 @ W_c2^T;  x4 = relu(relu(BN4(y4)) + x3); transpose
//   GEMM(p2): y5 = x4 @ W_p2^T;  store y5 [B,8] + stats5
// Branch-free lane padding: lanes n>=12 use clamped (in-bounds) load columns
// and zeroed BN coefficients, which force x3C/x4C to exactly 0 without any
// per-element exec-mask toggling in the hot loop.
// ---------------------------------------------------------------------------
__global__ __launch_bounds__(256)
void back_kernel(const float* __restrict__ y3,
                 const float* __restrict__ sums3, const float* __restrict__ g3,
                 const float* __restrict__ be3,
                 const float* __restrict__ wc, const float* __restrict__ bc,   // c2 [12,12]
                 const float* __restrict__ sums4, const float* __restrict__ g4,
                 const float* __restrict__ be4,
                 const float* __restrict__ wp, const float* __restrict__ bp,   // p2 [8,12]
                 float* __restrict__ y5, float* __restrict__ sums_out,
                 int numTiles, float invB) {
  __shared__ float xls[8 * TRW];
  const int lane = threadIdx.x & 31;
  const int half = lane >> 4;
  const int n    = lane & 15;
  const int ncl  = (n < 12) ? n : 11;   // clamped, always-in-bounds column
  float* myx = &xls[(threadIdx.x >> 5) * TRW];

  v2f bwc[3], bwp[3];
#pragma unroll
  for (int c = 0; c < 3; ++c) {
    const int kb = c * 4 + 2 * half;
    if (n < 12) { bwc[c].x = wc[n * 12 + kb]; bwc[c].y = wc[n * 12 + kb + 1]; }
    else        { bwc[c].x = 0.0f;            bwc[c].y = 0.0f; }
    if (n < 8)  { bwp[c].x = wp[n * 12 + kb]; bwp[c].y = wp[n * 12 + kb + 1]; }
    else        { bwp[c].x = 0.0f;            bwp[c].y = 0.0f; }
  }
  const float biasc = (n < 12) ? bc[n] : 0.0f;
  const float biasp = (n < 8)  ? bp[n] : 0.0f;

  // BN coefs computed with clamped index (in-bounds) and zeroed for pad lanes
  // so the inner loop needs no predication: fma(x, 0, 0) == 0.
  float s3C, t3C, s4C, t4C;
  bn_coef(sums3, g3, be3, ncl, invB, s3C, t3C);
  bn_coef(sums4, g4, be4, ncl, invB, s4C, t4C);
  if (n >= 12) { s3C = 0.0f; t3C = 0.0f; s4C = 0.0f; t4C = 0.0f; }

  const int wavesPerBlock = blockDim.x >> 5;
  const int wave = blockIdx.x * wavesPerBlock + (threadIdx.x >> 5);
  const int nw   = gridDim.x * wavesPerBlock;

  float ssum = 0.0f, ssq = 0.0f;
  for (int tile = wave; tile < numTiles; tile += nw) {
    const size_t rbase = (size_t)tile * 16 + 8 * half;

    // --- x3 = relu(BN3(y3)), C pattern (branch-free; pad lanes -> 0) ---
    float x3C[8];
#pragma unroll
    for (int r = 0; r < 8; ++r) {
      const float v = y3[(rbase + r) * 12 + ncl];
      x3C[r] = fmaxf(fmaf(v, s3C, t3C), 0.0f);
    }
    wave_sync_lds();
#pragma unroll
    for (int r = 0; r < 8; ++r) myx[(r + 8 * half) * TRS + n] = x3C[r];
    wave_sync_lds();

    // --- GEMM c2 ---
    v8f acc4;
#pragma unroll
    for (int r = 0; r < 8; ++r) acc4[r] = biasc;
#pragma unroll
    for (int c = 0; c < 3; ++c) {
      const int kb = c * 4 + 2 * half;
      const float2 xa = *(const float2*)&myx[n * TRS + kb];
      v2f a; a.x = xa.x; a.y = xa.y;
      acc4 = __builtin_amdgcn_wmma_f32_16x16x4_f32(false, a, false, bwc[c],
                                                   (short)0, acc4, false, false);
    }

    // --- x4 = relu(relu(BN4(y4)) + x3); branch-free; transpose ---
    float x4C[8];
#pragma unroll
    for (int r = 0; r < 8; ++r)
      x4C[r] = fmaxf(fmaxf(fmaf(acc4[r], s4C, t4C), 0.0f) + x3C[r], 0.0f);
    wave_sync_lds();
#pragma unroll
    for (int r = 0; r < 8; ++r) myx[(r + 8 * half) * TRS + n] = x4C[r];
    wave_sync_lds();

    // --- GEMM p2 ---
    v8f acc5;
#pragma unroll
    for (int r = 0; r < 8; ++r) acc5[r] = biasp;
#pragma unroll
    for (int c = 0; c < 3; ++c) {
      const int kb = c * 4 + 2 * half;
      const float2 xa = *(const float2*)&myx[n * TRS + kb];
      v2f a; a.x = xa.x; a.y = xa.y;
      acc5 = __builtin_amdgcn_wmma_f32_16x16x4_f32(false, a, false, bwp[c],
                                                   (short)0, acc5, false, false);
    }
    if (n < 8) {
#pragma unroll
      for (int r = 0; r < 8; ++r) {
        const float v = acc5[r];
        y5[(rbase + r) * 8 + n] = v;
        ssum += v; ssq += v * v;
      }
    }
  }

  ssum += __shfl_xor(ssum, 16, 32);
  ssq  += __shfl_xor(ssq, 16, 32);
  if (half == 0 && n < 8) {
    atomicAdd(&sums_out[n], ssum);
    atomicAdd(&sums_out[16 + n], ssq);
  }
}

// ---------------------------------------------------------------------------
// Final head: out = sigmoid(relu(BN5(y5)) . w_fc + b_fc)   (N=1 -> scalar)
// ---------------------------------------------------------------------------
__global__ __launch_bounds__(256)
void final_kernel(const float* __restrict__ y5, const float* __restrict__ sums5,
                  const float* __restrict__ g5, const float* __restrict__ be5,
                  const float* __restrict__ w_fc, const float* __restrict__ b_fc,
                  float* __restrict__ out, int B, float invB) {
  __shared__ float sc[8], sh[8], wv[8], bb[1];
  if (threadIdx.x < 8) {
    const int k = threadIdx.x;
    float s, t;
    bn_coef(sums5, g5, be5, k, invB, s, t);
    sc[k] = s; sh[k] = t; wv[k] = w_fc[k];
  }
  if (threadIdx.x == 0) bb[0] = b_fc[0];
  __syncthreads();

  int i = blockIdx.x * blockDim.x + threadIdx.x;
  const int stride = gridDim.x * blockDim.x;
  for (; i < B; i += stride) {
    const float4* p = (const float4*)(y5 + (size_t)i * 8);
    const float4 a = p[0];
    const float4 b = p[1];
    float acc = bb[0];
    acc += fmaxf(fmaf(a.x, sc[0], sh[0]), 0.0f) * wv[0];
    acc += fmaxf(fmaf(a.y, sc[1], sh[1]), 0.0f) * wv[1];
    acc += fmaxf(fmaf(a.z, sc[2], sh[2]), 0.0f) * wv[2];
    acc += fmaxf(fmaf(a.w, sc[3], sh[3]), 0.0f) * wv[3];
    acc += fmaxf(fmaf(b.x, sc[4], sh[4]), 0.0f) * wv[4];
    acc += fmaxf(fmaf(b.y, sc[5], sh[5]), 0.0f) * wv[5];
    acc += fmaxf(fmaf(b.z, sc[6], sh[6]), 0.0f) * wv[6];
    acc += fmaxf(fmaf(b.w, sc[7], sh[7]), 0.0f) * wv[7];
    out[i] = 1.0f / (1.0f + __expf(-acc));
  }
}

// ---------------------------------------------------------------------------
extern "C" void kernel_launch(void* const* d_in, const int* in_sizes, int n_in,
                              void* d_out, int out_size, void* d_ws, size_t ws_size,
                              hipStream_t stream) {
  const float* inp   = (const float*)d_in[0];
  const float* w_fm  = (const float*)d_in[1];
  const float* b_fm  = (const float*)d_in[2];
  const float* g_fm  = (const float*)d_in[3];
  const float* be_fm = (const float*)d_in[4];
  const float* w_c1  = (const float*)d_in[5];
  const float* b_c1  = (const float*)d_in[6];
  const float* g_c1  = (const float*)d_in[7];
  const float* be_c1 = (const float*)d_in[8];
  const float* w_p1  = (const float*)d_in[9];
  const float* b_p1  = (const float*)d_in[10];
  const float* g_p1  = (const float*)d_in[11];
  const float* be_p1 = (const float*)d_in[12];
  const float* w_c2  = (const float*)d_in[13];
  const float* b_c2  = (const float*)d_in[14];
  const float* g_c2  = (const float*)d_in[15];
  const float* be_c2 = (const float*)d_in[16];
  const float* w_p2  = (const float*)d_in[17];
  const float* b_p2  = (const float*)d_in[18];
  const float* g_p2  = (const float*)d_in[19];
  const float* be_p2 = (const float*)d_in[20];
  const float* w_fc  = (const float*)d_in[21];
  const float* b_fc  = (const float*)d_in[22];

  const int B        = in_sizes[0] / 8;   // 2,097,152
  const int numTiles = B / 16;
  const float invB   = 1.0f / (float)B;

  char* ws = (char*)d_ws;
  float* stats = (float*)ws;              // 5 stages x 32 floats
  float* y3 = (float*)(ws + 4096);        // [B,12] pre-BN   (96 MB)
  float* y5 = y3 + (size_t)B * 12;        // [B,8]  pre-BN   (64 MB)
  float* out = (float*)d_out;

  zero_stats_kernel<<<1, 256, 0, stream>>>(stats);

  const dim3 blk(256);
  const int G = 1024;  // 8192 waves, 16 row-tiles each

  // K1: stats1 of y1 = inp @ W_fm^T (no store)
  stage_kernel<8, 16, false, false><<<G, blk, 0, stream>>>(
      inp, nullptr, nullptr, nullptr, w_fm, b_fm, nullptr, stats + 0,
      numTiles, invB);

  // K2: recompute y1 -> x1 -> y2; stats2 (no store)
  front_kernel<false><<<G, blk, 0, stream>>>(
      inp, w_fm, b_fm, stats + 0, g_fm, be_fm, w_c1, b_c1,
      nullptr, nullptr, nullptr, nullptr, nullptr,
      nullptr, stats + 32, numTiles, invB);

  // K3: recompute y1,y2 -> x2 (residual) -> y3; store y3 + stats3
  front_kernel<true><<<G, blk, 0, stream>>>(
      inp, w_fm, b_fm, stats + 0, g_fm, be_fm, w_c1, b_c1,
      stats + 32, g_c1, be_c1, w_p1, b_p1,
      y3, stats + 64, numTiles, invB);

  // K4: stats4 of y4 = relu(BN3(y3)) @ W_c2^T (no store)
  stage_kernel<12, 12, true, false><<<G, blk, 0, stream>>>(
      y3, stats + 64, g_p1, be_p1, w_c2, b_c2, nullptr, stats + 96,
      numTiles, invB);

  // K5: recompute y4 -> x4 (residual) -> y5; store y5 + stats5
  back_kernel<<<G, blk, 0, stream>>>(
      y3, stats + 64, g_p1, be_p1, w_c2, b_c2,
      stats + 96, g_c2, be_c2, w_p2, b_p2,
      y5, stats + 128, numTiles, invB);

  // K6: out = sigmoid(relu(BN5(y5)) . w_fc + b_fc)
  final_kernel<<<4096, blk, 0, stream>>>(y5, stats + 128, g_p2, be_p2,
                                         w_fc, b_fc, out, B, invB);
}